// Candidate_Representation_50603304682318
// MI455X (gfx1250) — compile-verified
//
#include <hip/hip_runtime.h>
#include <math.h>

// ---------------------------------------------------------------------------
// Problem constants (from reference)
// ---------------------------------------------------------------------------
#define PP 128
#define LL 800
#define DD 200
#define HH 100
#define KK 2
#define MM (PP * KK)      // 256 candidates
#define MM1 (MM - 1)      // 255

typedef __attribute__((ext_vector_type(2))) float v2f;
typedef __attribute__((ext_vector_type(4))) float f4;
typedef __attribute__((ext_vector_type(8))) float v8f;

// d_out layout (flat float, concatenated in reference return order)
#define OFF_SCS 0LL
#define OFF_R   ((long long)MM * LL * DD)           // 40,960,000
#define OFF_ENC (OFF_R + (long long)MM * HH)        // +25,600
#define OFF_V   (OFF_ENC + (long long)MM * LL)      // +204,800
#define OFF_T   (OFF_V + (long long)MM * MM1)       // +65,280

// ---------------------------------------------------------------------------
// Kernel 1: S_Cs = S_p[p_idx] * span_mask    (bandwidth-dominated, 164MB out)
// one thread per float4 (D = 200 floats = 50 float4, 16B aligned rows)
// ---------------------------------------------------------------------------
__global__ void scs_kernel(const float* __restrict__ S_p,
                           const int* __restrict__ spans,
                           float* __restrict__ out) {
    long long idx = (long long)blockIdx.x * blockDim.x + threadIdx.x;
    const int V4 = DD / 4; // 50
    int v = (int)(idx % V4);
    int t = (int)((idx / V4) % LL);
    int m = (int)(idx / ((long long)V4 * LL));
    int s = spans[2 * m];
    int e = spans[2 * m + 1];
    int p = m >> 1;                      // p_idx = repeat(arange(P), K)
    f4 val = (f4){0.f, 0.f, 0.f, 0.f};
    if (t >= s && t <= e) {
        const f4* src = (const f4*)(S_p + ((long long)p * LL + t) * DD);
        val = src[v];
    }
    f4* dst = (f4*)(out + ((long long)m * LL + t) * DD);
    __builtin_nontemporal_store(val, dst + v);
}

// ---------------------------------------------------------------------------
// Kernel 2: encoded_candidates (256 x 800), written as float (ids < 2^24)
// ---------------------------------------------------------------------------
__global__ void enc_kernel(const int* __restrict__ spans,
                           const int* __restrict__ passages,
                           float* __restrict__ out_enc) {
    int idx = blockIdx.x * blockDim.x + threadIdx.x;
    if (idx >= MM * LL) return;
    int t = idx % LL;
    int m = idx / LL;
    int s = spans[2 * m];
    int e = spans[2 * m + 1];
    int p = m >> 1;
    int g = s + t;
    g = g < 0 ? 0 : (g > LL - 1 ? LL - 1 : g);
    int val = (t <= e - s) ? passages[p * LL + g] : 0;
    out_enc[idx] = (float)val;
}

// ---------------------------------------------------------------------------
// Kernel 3: r_Cs = tanh(sb@Wb^T + se@We^T)  via V_WMMA_F32_16X16X4_F32
// [sb|se] is 256x400 gathered on the fly from S_p at span start/end rows.
// One wave per 16x16 tile of the 256x(100->pad 112) output. 112 waves.
// A layout: lane<16 -> (row=lane, k=kk..kk+1); lane>=16 -> (row=lane-16, k=kk+2..kk+3)
// B layout mirrors A with rows<->cols. C/D: VGPR g -> M = g (+8 for lanes>=16).
// ---------------------------------------------------------------------------
__global__ void rcs_kernel(const float* __restrict__ S_p,
                           const int* __restrict__ spans,
                           const float* __restrict__ Wb,
                           const float* __restrict__ We,
                           float* __restrict__ r_out) {
    int wave = ((blockIdx.x * blockDim.x + threadIdx.x) >> 5); // 0..111
    int lane = threadIdx.x & 31;
    int m0 = (wave / 7) * 16;
    int h0 = (wave % 7) * 16;
    int row   = lane & 15;
    int khalf = (lane >> 4) << 1;        // 0 or 2

    int m = m0 + row;
    int s = spans[2 * m];
    int e = spans[2 * m + 1];
    int p = m >> 1;
    const float* sb = S_p + ((long long)p * LL + s) * DD;
    const float* se = S_p + ((long long)p * LL + e) * DD;

    int n = h0 + row;
    int nc = (n < HH) ? n : (HH - 1);    // clamp pad columns (outputs unstored)
    const float* wbr = Wb + nc * DD;
    const float* wer = We + nc * DD;

    v8f acc = (v8f){0.f, 0.f, 0.f, 0.f, 0.f, 0.f, 0.f, 0.f};
    for (int kk = 0; kk < DD; kk += 4) {
        v2f a = (v2f){sb[kk + khalf], sb[kk + khalf + 1]};
        v2f b = (v2f){wbr[kk + khalf], wbr[kk + khalf + 1]};
        acc = __builtin_amdgcn_wmma_f32_16x16x4_f32(false, a, false, b,
                                                    (short)0, acc, false, false);
    }
    for (int kk = 0; kk < DD; kk += 4) {
        v2f a = (v2f){se[kk + khalf], se[kk + khalf + 1]};
        v2f b = (v2f){wer[kk + khalf], wer[kk + khalf + 1]};
        acc = __builtin_amdgcn_wmma_f32_16x16x4_f32(false, a, false, b,
                                                    (short)0, acc, false, false);
    }

    int mbase = m0 + ((lane >> 4) << 3);
    int nn = h0 + (lane & 15);
    if (nn < HH) {
        #pragma unroll
        for (int g = 0; g < 8; ++g)
            r_out[(mbase + g) * HH + nn] = tanhf(acc[g]);
    }
}

// ---------------------------------------------------------------------------
// Kernel 4: c = r@Wc^T, o = r@Wo^T (blockIdx.z selects), 256x100 @ 100x100
// ---------------------------------------------------------------------------
__global__ void co_kernel(const float* __restrict__ r,
                          const float* __restrict__ Wc,
                          const float* __restrict__ Wo,
                          float* __restrict__ c_out,
                          float* __restrict__ o_out) {
    int wave = blockIdx.x * (blockDim.x >> 5) + (threadIdx.x >> 5); // 0..111
    int lane = threadIdx.x & 31;
    const float* Wsel = (blockIdx.z == 0) ? Wc : Wo;
    float* outp       = (blockIdx.z == 0) ? c_out : o_out;

    int m0 = (wave / 7) * 16;
    int h0 = (wave % 7) * 16;
    int row   = lane & 15;
    int khalf = (lane >> 4) << 1;

    const float* ar = r + (m0 + row) * HH;
    int n = h0 + row;
    int nc = (n < HH) ? n : (HH - 1);
    const float* br = Wsel + nc * HH;

    v8f acc = (v8f){0.f, 0.f, 0.f, 0.f, 0.f, 0.f, 0.f, 0.f};
    for (int kk = 0; kk < HH; kk += 4) {           // 100 % 4 == 0
        v2f a = (v2f){ar[kk + khalf], ar[kk + khalf + 1]};
        v2f b = (v2f){br[kk + khalf], br[kk + khalf + 1]};
        acc = __builtin_amdgcn_wmma_f32_16x16x4_f32(false, a, false, b,
                                                    (short)0, acc, false, false);
    }

    int mbase = m0 + ((lane >> 4) << 3);
    int nn = h0 + (lane & 15);
    if (nn < HH) {
        #pragma unroll
        for (int g = 0; g < 8; ++g)
            outp[(mbase + g) * HH + nn] = acc[g];
    }
}

// ---------------------------------------------------------------------------
// Kernel 5: V_full[m,n] = sum_h tanh(c[m,h]+o[n,h])*Wv[h]; write V (diag
// removed) and E = exp(V) into workspace. Block = m, thread = n.
// ---------------------------------------------------------------------------
__global__ void v_kernel(const float* __restrict__ c,
                         const float* __restrict__ o,
                         const float* __restrict__ Wv,
                         float* __restrict__ V_out,
                         float* __restrict__ E_ws) {
    int m = blockIdx.x;      // 0..255
    int n = threadIdx.x;     // 0..255
    __shared__ float csh[HH];
    __shared__ float wsh[HH];
    if (n < HH) {
        csh[n] = c[m * HH + n];
        wsh[n] = Wv[n];
    }
    __syncthreads();
    const float* orow = o + n * HH;
    float acc = 0.f;
    for (int h = 0; h < HH; ++h)
        acc += tanhf(csh[h] + orow[h]) * wsh[h];
    if (n != m) {
        int j = n - (n > m ? 1 : 0);
        V_out[m * MM1 + j] = acc;
        E_ws[m * MM1 + j]  = expf(acc);
    }
}

// ---------------------------------------------------------------------------
// Kernel 6: per-column (j of the M x (M-1) E matrix) reductions:
//   colE[j] = sum_m E[m,j];  colA[j] = sum_m E[m,j]/(colE[j]-E[m,j])
// ---------------------------------------------------------------------------
__global__ void col_kernel(const float* __restrict__ E_ws,
                           float* __restrict__ colE,
                           float* __restrict__ colA) {
    int j = blockIdx.x;      // 0..254
    int m = threadIdx.x;     // 0..255
    __shared__ float red[MM];
    __shared__ float ce_sh;
    float e = E_ws[m * MM1 + j];
    red[m] = e;
    __syncthreads();
    for (int s = MM / 2; s > 0; s >>= 1) {
        if (m < s) red[m] += red[m + s];
        __syncthreads();
    }
    if (m == 0) ce_sh = red[0];
    __syncthreads();
    float ce = ce_sh;
    float alpha = e / (ce - e);
    red[m] = alpha;
    __syncthreads();
    for (int s = MM / 2; s > 0; s >>= 1) {
        if (m < s) red[m] += red[m + s];
        __syncthreads();
    }
    if (m == 0) {
        colE[j] = ce;
        colA[j] = red[0];
    }
}

// ---------------------------------------------------------------------------
// Kernel 7: tilda[m,h] = sum_j (colA[j] - alpha[m,j]) * r[idx[m,j], h]
// Block = m; phase 1 builds s_rows[m,:] in LDS, phase 2 reduces over j.
// ---------------------------------------------------------------------------
__global__ void tilda_kernel(const float* __restrict__ E_ws,
                             const float* __restrict__ colE,
                             const float* __restrict__ colA,
                             const float* __restrict__ r,
                             float* __restrict__ t_out) {
    int m = blockIdx.x;      // 0..255
    __shared__ float ssh[MM1];
    for (int j = threadIdx.x; j < MM1; j += blockDim.x) {
        float e = E_ws[m * MM1 + j];
        float alpha = e / (colE[j] - e);
        ssh[j] = colA[j] - alpha;
    }
    __syncthreads();
    int h = threadIdx.x;
    if (h < HH) {
        float acc = 0.f;
        for (int j = 0; j < MM1; ++j) {
            int n = j + (j >= m ? 1 : 0);
            acc += ssh[j] * r[n * HH + h];
        }
        t_out[m * HH + h] = acc;
    }
}

// ---------------------------------------------------------------------------
// Launch
// ---------------------------------------------------------------------------
extern "C" void kernel_launch(void* const* d_in, const int* in_sizes, int n_in,
                              void* d_out, int out_size, void* d_ws, size_t ws_size,
                              hipStream_t stream) {
    const float* S_p      = (const float*)d_in[0];
    const int*   spans    = (const int*)  d_in[1];
    const int*   passages = (const int*)  d_in[2];
    const float* Wb       = (const float*)d_in[3];
    const float* We       = (const float*)d_in[4];
    const float* Wc       = (const float*)d_in[5];
    const float* Wo       = (const float*)d_in[6];
    const float* Wv       = (const float*)d_in[7];

    float* out = (float*)d_out;
    float* scs_o = out + OFF_SCS;
    float* r_o   = out + OFF_R;
    float* enc_o = out + OFF_ENC;
    float* v_o   = out + OFF_V;
    float* t_o   = out + OFF_T;

    float* ws   = (float*)d_ws;
    float* c_ws = ws;                      // 256*100
    float* o_ws = ws + MM * HH;            // 256*100
    float* E_ws = ws + 2 * MM * HH;        // 256*255
    float* colE = E_ws + MM * MM1;         // 255
    float* colA = colE + MM1;              // 255

    // 1) S_Cs: M*L*(D/4) float4 units = 10,240,000 -> 40,000 blocks x 256
    {
        long long total = (long long)MM * LL * (DD / 4);
        int blocks = (int)((total + 255) / 256);
        scs_kernel<<<blocks, 256, 0, stream>>>(S_p, spans, scs_o);
    }
    // 2) encoded candidates
    {
        int total = MM * LL;
        enc_kernel<<<(total + 255) / 256, 256, 0, stream>>>(spans, passages, enc_o);
    }
    // 3) r_Cs GEMM (112 waves = 14 blocks x 8 waves)
    rcs_kernel<<<14, 256, 0, stream>>>(S_p, spans, Wb, We, r_o);
    // 4) c and o GEMMs
    {
        dim3 grid(14, 1, 2);
        co_kernel<<<grid, 256, 0, stream>>>(r_o, Wc, Wo, c_ws, o_ws);
    }
    // 5) V + E
    v_kernel<<<MM, MM, 0, stream>>>(c_ws, o_ws, Wv, v_o, E_ws);
    // 6) column reductions
    col_kernel<<<MM1, MM, 0, stream>>>(E_ws, colE, colA);
    // 7) tilda_r_Cs
    tilda_kernel<<<MM, 128, 0, stream>>>(E_ws, colE, colA, r_o, t_o);
}